// SparseLinear_68092411511135
// MI455X (gfx1250) — compile-verified
//
#include <hip/hip_runtime.h>
#include <hip/hip_bf16.h>
#include <stdint.h>

#define BATCH   64
#define N_FEAT  50000
#define UNITS   1024

#define TPB      1024               // 32 waves of 32
#define NWAVES   (TPB / 32)
#define NWG      256
#define ACC_ELEMS (BATCH * UNITS)   // 65536 floats = 256 KB
#define ACC_BYTES (ACC_ELEMS * 4)
#define STAGE_BYTES_PER_WAVE 768    // 2 buffers * 3 arrays * 32 lanes * 4B
#define STAGE_ELEMS_PER_WAVE 192
#define SHMEM_BYTES (ACC_BYTES + NWAVES * STAGE_BYTES_PER_WAVE)  // 286720 <= 320KB

// ---------------------------------------------------------------------------
// CDNA5 async global->LDS copy (ASYNCcnt-tracked).  lds_addr is the 32-bit LDS
// byte address (low 32 bits of a generic pointer to __shared__), gaddr is the
// 64-bit global address.  Per ISA 15.18.3: GLOBAL_LOAD_ASYNC_TO_LDS_B32.
// ---------------------------------------------------------------------------
__device__ __forceinline__ void async_copy_b32_to_lds(uint32_t lds_addr,
                                                      unsigned long long gaddr) {
    asm volatile("global_load_async_to_lds_b32 %0, %1, off"
                 :
                 : "v"(lds_addr), "v"(gaddr)
                 : "memory");
}
__device__ __forceinline__ void wait_asynccnt_le3() {
    asm volatile("s_wait_asynccnt 0x3" ::: "memory");
}
__device__ __forceinline__ void wait_asynccnt_0() {
    asm volatile("s_wait_asynccnt 0x0" ::: "memory");
}

// ---------------------------------------------------------------------------
// Kernel 0: zero the output (harness poisons d_out with 0xAA).
// ---------------------------------------------------------------------------
__global__ void zero_kernel(float* __restrict__ p, int n) {
    int i = blockIdx.x * blockDim.x + threadIdx.x;
    if (i < n) p[i] = 0.0f;
}

// ---------------------------------------------------------------------------
// Kernel 1: transpose x[64, 50000] -> xT[50000, 64] so each feature's 64-batch
// vector is one contiguous 256B row (the gather unit of the main kernel).
// ---------------------------------------------------------------------------
__global__ void __launch_bounds__(256) transpose_kernel(const float* __restrict__ x,
                                                        float* __restrict__ xT) {
    __shared__ float tile[64][65];  // +1 pad: stride 65 is coprime with 64 banks
    const int f0 = blockIdx.x * 64;
    const int tx = threadIdx.x & 63;  // 0..63
    const int tq = threadIdx.x >> 6;  // 0..3
    for (int b = tq; b < 64; b += 4) {
        int f = f0 + tx;
        tile[b][tx] = (f < N_FEAT) ? x[(size_t)b * N_FEAT + f] : 0.0f;
    }
    __syncthreads();
    for (int fr = tq; fr < 64; fr += 4) {
        int f = f0 + fr;
        if (f < N_FEAT) xT[(size_t)f * 64 + tx] = tile[tx][fr];
    }
}

// ---------------------------------------------------------------------------
// Kernel 2: main scatter.  Each WG owns a full 64x1024 f32 accumulator in LDS
// (256 KB of the WGP's 320 KB), processes a contiguous nnz chunk, then merges
// with global_atomic_add_f32.  Triplet streams are double-buffered through LDS
// with async copies (ASYNCcnt).
// ---------------------------------------------------------------------------
__global__ void __launch_bounds__(TPB, 1)
spmm_kernel(const float* __restrict__ xT, const float* __restrict__ x,
            const float* __restrict__ w, const int* __restrict__ row_idx,
            const int* __restrict__ col_idx, float* __restrict__ out,
            int nnz, int use_xt) {
    extern __shared__ float lds[];

    const int tid  = threadIdx.x;
    const int lane = tid & 31;
    const int wave = tid >> 5;

    // --- zero the accumulator -------------------------------------------------
    for (int i = tid; i < ACC_ELEMS; i += TPB) lds[i] = 0.0f;
    __syncthreads();

    // --- this WG's nnz chunk --------------------------------------------------
    const long long per = ((long long)nnz + gridDim.x - 1) / (long long)gridDim.x;
    long long c0 = (long long)blockIdx.x * per;
    long long c1 = c0 + per;
    if (c0 > nnz) c0 = nnz;
    if (c1 > nnz) c1 = nnz;
    const int k0 = (int)c0, k1 = (int)c1;

    // --- per-wave staging buffers (beyond the accumulator) --------------------
    float* stage = lds + ACC_ELEMS + wave * STAGE_ELEMS_PER_WAVE;  // 2 x 96 floats

    auto issue_chunk = [&](int kstart, int buf) {
        int kk = kstart + lane;
        kk = (kk < nnz) ? kk : (nnz - 1);
        kk = (kk > 0) ? kk : 0;
        float* dstp = stage + buf * 96 + lane;
        uint32_t dst = (uint32_t)(size_t)dstp;  // LDS byte address (low 32 bits)
        async_copy_b32_to_lds(dst,       (unsigned long long)(const void*)(row_idx + kk));
        async_copy_b32_to_lds(dst + 128, (unsigned long long)(const void*)(col_idx + kk));
        async_copy_b32_to_lds(dst + 256, (unsigned long long)(const void*)(w + kk));
    };

    const int stride = NWAVES * 32;
    const int kfirst = k0 + wave * 32;
    int buf = 0;

    issue_chunk(kfirst, 0);  // prologue (clamped, harmless if wave has no work)

    for (int kc = kfirst; kc < k1; kc += stride) {
        issue_chunk(kc + stride, buf ^ 1);  // prefetch next chunk
        wait_asynccnt_le3();                // current buffer's 3 copies done

        const float* sp = stage + buf * 96 + lane;
        int   r  = ((const int*)sp)[0];
        int   c  = ((const int*)(sp + 32))[0];
        float ww = sp[64];
        ww = ((kc + lane) < k1) ? ww : 0.0f;  // tail lanes contribute zero

        const int b0 = lane * 2;  // each lane covers batches 2*lane, 2*lane+1
#pragma unroll
        for (int j = 0; j < 32; ++j) {
            int   rj = __builtin_amdgcn_readlane(r, j);
            int   cj = __builtin_amdgcn_readlane(c, j);
            float wj = __uint_as_float(
                __builtin_amdgcn_readlane(__float_as_uint(ww), j));
            float x0, x1;
            if (use_xt) {
                float2 xv = *(const float2*)(xT + (size_t)rj * 64 + b0);
                x0 = xv.x; x1 = xv.y;
            } else {
                x0 = x[(size_t)b0 * N_FEAT + rj];
                x1 = x[(size_t)(b0 + 1) * N_FEAT + rj];
            }
            // bank-conflict-free ds_add_f32: lane l -> banks 2l and 2l+1
            atomicAdd(&lds[cj * 64 + b0],     wj * x0);
            atomicAdd(&lds[cj * 64 + b0 + 1], wj * x1);
        }
        buf ^= 1;
    }

    wait_asynccnt_0();  // drain outstanding async copies before reuse/exit
    __syncthreads();

    // --- merge partials into global out[b, u] ---------------------------------
    for (int i = tid; i < ACC_ELEMS; i += TPB) {
        float v = lds[i];
        if (v != 0.0f) {
            int col = i >> 6, b = i & 63;
            unsafeAtomicAdd(out + (size_t)b * UNITS + col, v);  // global_atomic_add_f32
        }
    }
}

// ---------------------------------------------------------------------------
// Host launcher.  Inputs (setup_inputs order): x, w, row_idx, col_idx.
// ---------------------------------------------------------------------------
extern "C" void kernel_launch(void* const* d_in, const int* in_sizes, int n_in,
                              void* d_out, int out_size, void* d_ws, size_t ws_size,
                              hipStream_t stream) {
    const float* x       = (const float*)d_in[0];
    const float* w       = (const float*)d_in[1];
    const int*   row_idx = (const int*)d_in[2];
    const int*   col_idx = (const int*)d_in[3];
    float*       out     = (float*)d_out;
    const int    nnz     = in_sizes[1];

    const size_t xt_bytes = (size_t)N_FEAT * BATCH * sizeof(float);  // 12.8 MB
    const int    use_xt   = (ws_size >= xt_bytes) ? 1 : 0;
    float*       xT       = (float*)d_ws;

    zero_kernel<<<(out_size + 255) / 256, 256, 0, stream>>>(out, out_size);

    if (use_xt) {
        transpose_kernel<<<(N_FEAT + 63) / 64, 256, 0, stream>>>(x, xT);
    }

    // Allow >64KB dynamic LDS (CDNA5 WGP: up to 320KB per workgroup).
    (void)hipFuncSetAttribute((const void*)spmm_kernel,
                              hipFuncAttributeMaxDynamicSharedMemorySize,
                              (int)SHMEM_BYTES);

    spmm_kernel<<<NWG, TPB, SHMEM_BYTES, stream>>>(xT, x, w, row_idx, col_idx,
                                                   out, nnz, use_xt);
}